// Otimizador_36850819400596
// MI455X (gfx1250) — compile-verified
//
#include <hip/hip_runtime.h>

#define BB 4096
#define DD 512
#define KK 128
#define AA 32
#define PP 16

typedef __attribute__((ext_vector_type(2))) float v2f;
typedef __attribute__((ext_vector_type(8))) float v8f;

__device__ __forceinline__ v8f wmma4(v2f a, v2f b, v8f c) {
  // D = A(16x4 f32) * B(4x16 f32) + C(16x16 f32)
  return __builtin_amdgcn_wmma_f32_16x16x4_f32(false, a, false, b, (short)0, c,
                                               false, false);
}

// ---------------------------------------------------------------------------
// k0: M = W @ W^T [32x32], mumu[k] = alpha_k M alpha_k^T  (tiny, one block)
// ---------------------------------------------------------------------------
__global__ void k0_prep(const float* __restrict__ alpha, const float* __restrict__ W,
                        float* __restrict__ Mout, float* __restrict__ mumu) {
  __shared__ float Ms[AA * AA];
  int t = threadIdx.x;
  for (int e = t; e < AA * AA; e += blockDim.x) {
    int i = e >> 5, j = e & 31;
    float acc = 0.f;
    for (int d = 0; d < DD; ++d) acc += W[i * DD + d] * W[j * DD + d];
    Ms[e] = acc;
    Mout[e] = acc;
  }
  __syncthreads();
  for (int k = t; k < KK; k += blockDim.x) {
    float acc = 0.f;
    for (int i = 0; i < AA; ++i) {
      float s = 0.f;
      for (int j = 0; j < AA; ++j) s += Ms[i * AA + j] * alpha[k * AA + j];
      acc += alpha[k * AA + i] * s;
    }
    mumu[k] = acc;
  }
}

// ---------------------------------------------------------------------------
// k1: Y = thetas @ W^T [4096,32] via WMMA f32; fused tt[b] = ||theta_b||^2
// wave tile: 16 rows x 32 cols, K-loop over D=512 in steps of 4
// ---------------------------------------------------------------------------
__global__ void k1_Y_tt(const float* __restrict__ thetas, const float* __restrict__ W,
                        float* __restrict__ Y, float* __restrict__ tt) {
  int wave = threadIdx.x >> 5;
  int lane = threadIdx.x & 31;
  int tile = blockIdx.x * 8 + wave;   // 0..255
  int b0 = tile * 16;
  int m = lane & 15;
  int h = lane >> 4;                  // lane half selects K pair {2h,2h+1} mod 4
  const float* arow = thetas + (size_t)(b0 + m) * DD + 2 * h;
  const float* w1 = W + (size_t)m * DD + 2 * h;
  const float* w2 = W + (size_t)(16 + m) * DD + 2 * h;
  v8f c0 = {}, c1 = {};
  float tta = 0.f;
  for (int kk = 0; kk < DD; kk += 4) {
    v2f a  = *(const v2f*)(arow + kk);
    v2f b1 = *(const v2f*)(w1 + kk);
    v2f b2 = *(const v2f*)(w2 + kk);
    c0 = wmma4(a, b1, c0);
    c1 = wmma4(a, b2, c1);
    tta += a.x * a.x + a.y * a.y;     // each theta element touched exactly once
  }
  float tto = __shfl_xor(tta, 16);
  if (h == 0) tt[b0 + m] = tta + tto;
  #pragma unroll
  for (int r = 0; r < 8; ++r) {       // C layout: reg r -> M = r + 8*h, N = m
    int row = b0 + r + 8 * h;
    Y[(size_t)row * AA + m]      = c0[r];
    Y[(size_t)row * AA + 16 + m] = c1[r];
  }
}

// ---------------------------------------------------------------------------
// k2: X1 = Y @ alpha^T [16,128] per wave, then rowwise softmax -> w,
//     c = w@alpha, cMc, tdot -> per-sample loss part (score-norm term)
// ---------------------------------------------------------------------------
__global__ void k2_softmax(const float* __restrict__ Y, const float* __restrict__ alpha,
                           const float* __restrict__ mumu, const float* __restrict__ Mmat,
                           const float* __restrict__ tt, const int* __restrict__ Np,
                           float* __restrict__ wbuf, float* __restrict__ loss1) {
  __shared__ float X1s[8][16 * 128];  // 64 KB, one 16x128 tile per wave
  int wave = threadIdx.x >> 5;
  int lane = threadIdx.x & 31;
  int tile = blockIdx.x * 8 + wave;
  int b0 = tile * 16;
  int m = lane & 15;
  int h = lane >> 4;
  const float* arow = Y + (size_t)(b0 + m) * AA + 2 * h;
  v8f zero = {};
  v8f acc[8];
  #pragma unroll
  for (int c = 0; c < 8; ++c) acc[c] = zero;
  for (int kk = 0; kk < AA; kk += 4) {
    v2f a = *(const v2f*)(arow + kk);
    #pragma unroll
    for (int c = 0; c < 8; ++c) {
      v2f b = *(const v2f*)(alpha + (size_t)(16 * c + m) * AA + kk + 2 * h);
      acc[c] = wmma4(a, b, acc[c]);
    }
  }
  #pragma unroll
  for (int c = 0; c < 8; ++c)
    #pragma unroll
    for (int r = 0; r < 8; ++r)
      X1s[wave][(r + 8 * h) * KK + 16 * c + m] = acc[c][r];
  __syncthreads();

  if (h == 0) {                        // 16 lanes each own one row of the tile
    int b = b0 + m;
    const float* xr = &X1s[wave][m * KK];
    float Nf = (float)(*Np);
    float temper = Nf / (Nf + 1.0f);
    float mx = -1e30f;
    for (int k = 0; k < KK; ++k) mx = fmaxf(mx, xr[k] - 0.5f * mumu[k]);
    float ssum = 0.f;
    for (int k = 0; k < KK; ++k) ssum += expf(xr[k] - 0.5f * mumu[k] - mx);
    float inv = 1.0f / ssum;
    float tdot = 0.f;
    float cv[AA];
    #pragma unroll
    for (int a = 0; a < AA; ++a) cv[a] = 0.f;
    for (int k = 0; k < KK; ++k) {
      float wk = expf(xr[k] - 0.5f * mumu[k] - mx) * inv;
      wbuf[(size_t)b * KK + k] = wk;
      tdot += wk * xr[k];
      const float* ak = alpha + (size_t)k * AA;
      #pragma unroll
      for (int a = 0; a < AA; ++a) cv[a] += wk * ak[a];
    }
    float cMc = 0.f;
    for (int i = 0; i < AA; ++i) {
      float s = 0.f;
      const float* Mi = Mmat + i * AA;
      #pragma unroll
      for (int j = 0; j < AA; ++j) s += Mi[j] * cv[j];
      cMc += cv[i] * s;
    }
    // 0.5*||g||^2 - temper*D   (the -D part of the Hutchinson trace folded in)
    loss1[b] = 0.5f * temper * temper * (cMc - 2.f * tdot + tt[b])
             - temper * (float)DD;
  }
}

// ---------------------------------------------------------------------------
// k3 (dominant): stream v_int once.  Per wave tile of 16 (p,b)-rows:
//   U = (2v-1) @ W^T  [16,32]   (128 k-steps x 2 WMMA)
//   S2 = U @ alpha^T  [16,128]  (8 k-steps x 8 WMMA, U repacked via LDS)
//   contrib = sum_k w*S2^2 - (sum_k w*S2)^2  per row; deterministic block sum
// ---------------------------------------------------------------------------
__global__ void k3_main(const int* __restrict__ vint, const float* __restrict__ W,
                        const float* __restrict__ alpha, const float* __restrict__ wbuf,
                        float* __restrict__ partials) {
  __shared__ float sbuf[8][2048];     // 64 KB, per-wave scratch (U then S2)
  int wave = threadIdx.x >> 5;
  int lane = threadIdx.x & 31;
  int tile = blockIdx.x * 8 + wave;   // 0..4095
  int r0 = tile * 16;                 // global (p,b) row base
  int m = lane & 15;
  int h = lane >> 4;
  float* Us = sbuf[wave];

  // phase 1: U accumulate
  const int*   vrow = vint + (size_t)(r0 + m) * DD + 2 * h;
  const float* w1 = W + (size_t)m * DD + 2 * h;
  const float* w2 = W + (size_t)(16 + m) * DD + 2 * h;
  v8f u0 = {}, u1 = {};
  for (int kk = 0; kk < DD; kk += 4) {
    int2 iv = *(const int2*)(vrow + kk);
    v2f a;
    a.x = (float)(2 * iv.x - 1);      // Rademacher, exact in f32
    a.y = (float)(2 * iv.y - 1);
    v2f b1 = *(const v2f*)(w1 + kk);
    v2f b2 = *(const v2f*)(w2 + kk);
    u0 = wmma4(a, b1, u0);
    u1 = wmma4(a, b2, u1);
  }
  // phase 2: dump U tile [16][32] (C layout -> row major)
  #pragma unroll
  for (int r = 0; r < 8; ++r) {
    Us[(r + 8 * h) * AA + m]      = u0[r];
    Us[(r + 8 * h) * AA + 16 + m] = u1[r];
  }
  __syncthreads();
  // phase 3: S2 = U @ alpha^T
  v8f zero = {};
  v8f s2[8];
  #pragma unroll
  for (int c = 0; c < 8; ++c) s2[c] = zero;
  for (int kk = 0; kk < AA; kk += 4) {
    v2f a = *(const v2f*)(Us + m * AA + kk + 2 * h);   // A-frag from U tile
    #pragma unroll
    for (int c = 0; c < 8; ++c) {
      v2f b = *(const v2f*)(alpha + (size_t)(16 * c + m) * AA + kk + 2 * h);
      s2[c] = wmma4(a, b, s2[c]);
    }
  }
  __syncthreads();
  // phase 4: dump S2 tile [16][128] over same per-wave region
  #pragma unroll
  for (int c = 0; c < 8; ++c)
    #pragma unroll
    for (int r = 0; r < 8; ++r)
      Us[(r + 8 * h) * KK + 16 * c + m] = s2[c][r];
  __syncthreads();
  // phase 5: per-row weighted moments
  float contrib = 0.f;
  if (h == 0) {
    int b = (r0 + m) & (BB - 1);
    const float* wr = wbuf + (size_t)b * KK;
    const float* sr = Us + m * KK;
    float q1 = 0.f, q2 = 0.f;
    for (int k = 0; k < KK; ++k) {
      float s = sr[k], wk = wr[k];
      q1 += wk * s;
      q2 += wk * s * s;
    }
    contrib = q2 - q1 * q1;
  }
  __syncthreads();
  float* redbuf = &sbuf[0][0];        // reuse LDS for deterministic reduction
  redbuf[threadIdx.x] = contrib;
  __syncthreads();
  if (threadIdx.x == 0) {
    float s = 0.f;
    for (int i = 0; i < 256; ++i) s += redbuf[i];
    partials[blockIdx.x] = s;
  }
}

// ---------------------------------------------------------------------------
// k4: final deterministic reduction -> scalar loss
// ---------------------------------------------------------------------------
__global__ void k4_final(const float* __restrict__ loss1, const float* __restrict__ partials,
                         const int* __restrict__ Np, float* __restrict__ out) {
  __shared__ float red[256];
  int t = threadIdx.x;
  float s1 = 0.f;
  for (int i = t; i < BB; i += 256) s1 += loss1[i];
  float s2 = 0.f;
  for (int i = t; i < 512; i += 256) s2 += partials[i];
  red[t] = s1;
  __syncthreads();
  for (int off = 128; off > 0; off >>= 1) {
    if (t < off) red[t] += red[t + off];
    __syncthreads();
  }
  float tot1 = red[0];
  __syncthreads();
  red[t] = s2;
  __syncthreads();
  for (int off = 128; off > 0; off >>= 1) {
    if (t < off) red[t] += red[t + off];
    __syncthreads();
  }
  if (t == 0) {
    float Nf = (float)(*Np);
    float temper = Nf / (Nf + 1.0f);
    out[0] = tot1 / (float)BB + temper * red[0] / ((float)PP * (float)BB);
  }
}

extern "C" void kernel_launch(void* const* d_in, const int* in_sizes, int n_in,
                              void* d_out, int out_size, void* d_ws, size_t ws_size,
                              hipStream_t stream) {
  (void)in_sizes; (void)n_in; (void)out_size; (void)ws_size;
  const float* thetas = (const float*)d_in[0];
  const float* alpha  = (const float*)d_in[1];
  const float* W      = (const float*)d_in[2];
  const int*   vint   = (const int*)d_in[3];
  const int*   Np     = (const int*)d_in[4];
  // d_in[5] (K_alpha_samples): log K shifts logp only; loss uses grad/Hessian, so unused.

  float* ws       = (float*)d_ws;
  float* M        = ws;               // 1024
  float* mumu     = M + 1024;         // 128
  float* tt       = mumu + 128;       // 4096
  float* Y        = tt + 4096;        // 4096*32
  float* wbuf     = Y + 4096 * 32;    // 4096*128
  float* loss1    = wbuf + 4096 * 128;// 4096
  float* partials = loss1 + 4096;     // 512
  float* outf     = (float*)d_out;

  k0_prep   <<<1,   256, 0, stream>>>(alpha, W, M, mumu);
  k1_Y_tt   <<<32,  256, 0, stream>>>(thetas, W, Y, tt);
  k2_softmax<<<32,  256, 0, stream>>>(Y, alpha, mumu, M, tt, Np, wbuf, loss1);
  k3_main   <<<512, 256, 0, stream>>>(vint, W, alpha, wbuf, partials);
  k4_final  <<<1,   256, 0, stream>>>(loss1, partials, Np, outf);
}